// MaskedPredictionModule_38439957299574
// MI455X (gfx1250) — compile-verified
//
#include <hip/hip_runtime.h>
#include <math.h>

#define BATCH   32
#define HDIM    64
#define WDIM    64
#define FH      1024
#define FW      1024
#define HW      (HDIM * WDIM)        // 4096
#define NPATCH  (BATCH * HW)         // 131072

typedef float v2f __attribute__((ext_vector_type(2)));
typedef float v8f __attribute__((ext_vector_type(8)));
typedef int   vi4 __attribute__((ext_vector_type(4)));

// ---- CDNA5 async copy: global -> LDS (16B per lane), tracked by ASYNCcnt ----
__device__ __forceinline__ void async_copy16(const float* gsrc, float* ldst) {
  __builtin_amdgcn_global_load_async_to_lds_b128(
      (__attribute__((address_space(1))) vi4*)gsrc,
      (__attribute__((address_space(3))) vi4*)ldst,
      /*offset=*/0, /*cpol=*/0);
}
__device__ __forceinline__ void wait_async_le3() {
  asm volatile("s_wait_asynccnt 0x3" ::: "memory");
}
__device__ __forceinline__ void wait_async_le0() {
  asm volatile("s_wait_asynccnt 0x0" ::: "memory");
}
__device__ __forceinline__ void wait_ds0() {
  asm volatile("s_wait_dscnt 0x0" ::: "memory");
}

// ============================================================================
// Kernel 0: zero the comb counter (scratch must be re-inited every launch)
// ============================================================================
__global__ void init_kernel(int* comb_count) { *comb_count = 0; }

// ============================================================================
// Kernel 1: stream both masks once, 16x16 avg-pool + threshold.
// One workgroup (256 thr) per (batch, block-row): a 16-row x 1024-col strip.
// Rows stream through LDS via async b128 copies, 4-deep pipeline.
// Lane t owns columns 4t..4t+3 -> pooled column t/4.
// ============================================================================
__global__ void pool_masks_kernel(const float* __restrict__ needle,
                                  const float* __restrict__ prostate,
                                  unsigned char* __restrict__ nm_out,
                                  unsigned char* __restrict__ comb_out,
                                  int* __restrict__ comb_count) {
  __shared__ __align__(16) float buf[4][FW];   // 16 KB, 4-row ring
  const int t  = threadIdx.x;                  // 0..255
  const int wg = blockIdx.x;                   // 0..2047
  const int b  = wg >> 6;                      // batch
  const int r  = wg & 63;                      // pooled row (block row)
  const size_t strip = ((size_t)b * FH + (size_t)r * 16) * FW;

  float sums[2];
  #pragma unroll
  for (int m = 0; m < 2; ++m) {
    const float* src = (m == 0 ? needle : prostate) + strip;
    // prologue: rows 0..3 in flight
    #pragma unroll
    for (int j = 0; j < 4; ++j)
      async_copy16(src + (size_t)j * FW + t * 4, &buf[j][t * 4]);

    float acc = 0.0f;
    for (int i = 0; i < 16; ++i) {
      if (i < 12) wait_async_le3();  // oldest of 4 in flight is done
      else        wait_async_le0();  // tail: drain
      float4 v = *(const float4*)(&buf[i & 3][t * 4]);
      acc += (v.x + v.y) + (v.z + v.w);
      wait_ds0();                    // LDS read done before ring slot is reused
      if (i + 4 < 16)
        async_copy16(src + (size_t)(i + 4) * FW + t * 4, &buf[(i + 4) & 3][t * 4]);
    }
    sums[m] = acc;
  }

  // group-of-4 lane reduce completes the 16x16 block sum
  float s0 = sums[0], s1 = sums[1];
  s0 += __shfl_xor(s0, 1, 32);  s0 += __shfl_xor(s0, 2, 32);
  s1 += __shfl_xor(s1, 1, 32);  s1 += __shfl_xor(s1, 2, 32);

  const bool nm   = s0 > 128.0f;       // mean > 0.5  <=>  sum > 128
  const bool pm   = s1 > 128.0f;
  const bool comb = nm && pm;
  const bool writer = ((t & 3) == 0);

  if (writer) {
    const int idx = ((b << 6) + r) * 64 + (t >> 2);   // b*4096 + r*64 + pooled col
    nm_out[idx]   = nm   ? 1 : 0;
    comb_out[idx] = comb ? 1 : 0;
  }
  unsigned long long bal = __ballot(writer && comb);
  if ((t & 31) == 0) {
    int c = (int)__popcll(bal);
    if (c) atomicAdd(comb_count, c);
  }
}

// ============================================================================
// Kernel 2: elementwise patch outputs (bandwidth-trivial)
// ============================================================================
__global__ void patch_kernel(const float* __restrict__ logits,
                             const int* __restrict__ label,
                             const unsigned char* __restrict__ nm,
                             const unsigned char* __restrict__ comb,
                             const int* __restrict__ comb_count,
                             float* __restrict__ out) {
  const int i = blockIdx.x * blockDim.x + threadIdx.x;   // 0..131071
  const bool use_comb = (*comb_count) > 0;
  const unsigned char mb = use_comb ? comb[i] : nm[i];
  const float w = mb ? 1.0f : 0.0f;
  const float x = logits[i];
  const float p = 1.0f / (1.0f + __expf(-x));
  const float lab = (float)label[i >> 12];

  float* patch_pred  = out + 64;
  float* patch_logit = out + 64 + NPATCH;
  float* patch_lab   = out + 64 + 2 * NPATCH;
  float* mflat       = out + 64 + 3 * NPATCH;
  patch_pred[i]  = p * w;
  patch_logit[i] = x * w;
  patch_lab[i]   = lab * w;
  mflat[i]       = w;
}

// ============================================================================
// Kernel 3: core predictions via V_WMMA_F32_16X16X4_F32.
// Wave w handles batches 16w..16w+15. Each lane pre-reduces two 1024-patch
// chunks into its two A-matrix VGPRs (A is 16x4 f32: lanes 0-15 hold K0/K1,
// lanes 16-31 hold K2/K3 for the same 16 batches). D = A x ones gives row
// sums D[m][*] = sum_k A[m][k], i.e. the full 4096-patch sum per batch.
// ============================================================================
__global__ void core_kernel(const float* __restrict__ logits,
                            const int* __restrict__ label,
                            const unsigned char* __restrict__ nm,
                            const unsigned char* __restrict__ comb,
                            const int* __restrict__ comb_count,
                            float* __restrict__ out) {
  const int t    = threadIdx.x;       // 0..63 (2 waves)
  const int wave = t >> 5;
  const int lane = t & 31;
  const int base = wave * 16;                 // batch base for this wave
  const int bb   = base + (lane & 15);        // batch owned by this lane
  const int ch0  = (lane < 16) ? 0 : 2;       // K-chunks 0,1 or 2,3

  const bool use_comb = (*comb_count) > 0;
  const unsigned char* msk = use_comb ? comb : nm;

  v2f ap, aw;
  #pragma unroll
  for (int c = 0; c < 2; ++c) {
    const size_t off = (size_t)bb * HW + (size_t)(ch0 + c) * 1024;
    float sp = 0.0f, sw = 0.0f;
    for (int j = 0; j < 1024; j += 4) {
      float4 x = *(const float4*)(logits + off + j);
      uchar4 m = *(const uchar4*)(msk + off + j);
      float w0 = m.x ? 1.0f : 0.0f, w1 = m.y ? 1.0f : 0.0f;
      float w2 = m.z ? 1.0f : 0.0f, w3 = m.w ? 1.0f : 0.0f;
      sp += w0 / (1.0f + __expf(-x.x)) + w1 / (1.0f + __expf(-x.y))
          + w2 / (1.0f + __expf(-x.z)) + w3 / (1.0f + __expf(-x.w));
      sw += (w0 + w1) + (w2 + w3);
    }
    ap[c] = sp;  aw[c] = sw;
  }

  v8f zero = {};
  v2f ones; ones[0] = 1.0f; ones[1] = 1.0f;
  // D = A x ones + 0  ->  per-batch totals, replicated across all 16 columns
  v8f dp = __builtin_amdgcn_wmma_f32_16x16x4_f32(false, ap, false, ones,
                                                 (short)0, zero, false, false);
  v8f dw = __builtin_amdgcn_wmma_f32_16x16x4_f32(false, aw, false, ones,
                                                 (short)0, zero, false, false);

  // C/D layout: VGPR r = row M=r (lanes 0-15) / M=8+r (lanes 16-31)
  if (lane == 0 || lane == 16) {
    const int b0 = base + (lane == 0 ? 0 : 8);
    #pragma unroll
    for (int rI = 0; rI < 8; ++rI)
      out[b0 + rI] = dp[rI] / fmaxf(dw[rI], 1.0f);   // core_predictions
  }
  if (t < 32) out[32 + t] = (float)label[t];         // core_labels
}

// ============================================================================
extern "C" void kernel_launch(void* const* d_in, const int* in_sizes, int n_in,
                              void* d_out, int out_size, void* d_ws, size_t ws_size,
                              hipStream_t stream) {
  const float* logits   = (const float*)d_in[0];   // (32,1,64,64)
  const float* needle   = (const float*)d_in[1];   // (32,1,1024,1024)
  const float* prostate = (const float*)d_in[2];   // (32,1,1024,1024)
  const int*   label    = (const int*)d_in[3];     // (32,)
  float* out = (float*)d_out;

  int* comb_count        = (int*)d_ws;
  unsigned char* nm_ws   = (unsigned char*)d_ws + 128;
  unsigned char* comb_ws = nm_ws + NPATCH;

  hipLaunchKernelGGL(init_kernel, dim3(1), dim3(1), 0, stream, comb_count);
  hipLaunchKernelGGL(pool_masks_kernel, dim3(BATCH * HDIM), dim3(256), 0, stream,
                     needle, prostate, nm_ws, comb_ws, comb_count);
  hipLaunchKernelGGL(patch_kernel, dim3(NPATCH / 256), dim3(256), 0, stream,
                     logits, label, nm_ws, comb_ws, comb_count, out);
  hipLaunchKernelGGL(core_kernel, dim3(1), dim3(64), 0, stream,
                     logits, label, nm_ws, comb_ws, comb_count, out);
}